// GCNNet_15358803050970
// MI455X (gfx1250) — compile-verified
//
#include <hip/hip_runtime.h>

typedef __attribute__((ext_vector_type(2))) float v2f;
typedef __attribute__((ext_vector_type(8))) float v8f;

static constexpr int BATCH = 32;
static constexpr int NPTS  = 2048;
static constexpr int KNN   = 8;
static constexpr float NEG_SLOPE = 0.2f;
static constexpr float FMAX = 3.402823466e38f;

// --------------------------------------------------------------------------
// KNN: reference does xyz_r = reshape(xyz,(B,3,N)) (a *reshape*, not a
// transpose), so "point m" coords are buf[m], buf[m+2048], buf[m+4096] of the
// flat per-batch (N,3) buffer. Stage the whole batch cloud in LDS (32 KB).
// --------------------------------------------------------------------------
__global__ __launch_bounds__(256) void knn_kernel(const float* __restrict__ xyz,
                                                  int* __restrict__ idx)
{
    __shared__ float sx[NPTS], sy[NPTS], sz[NPTS], sxx[NPTS];
    const int b     = blockIdx.x >> 3;
    const int chunk = blockIdx.x & 7;
    const float* base = xyz + (size_t)b * NPTS * 3;
    for (int m = threadIdx.x; m < NPTS; m += 256) {
        float x = base[m];
        float y = base[NPTS + m];
        float z = base[2 * NPTS + m];
        sx[m] = x; sy[m] = y; sz[m] = z;
        sxx[m] = x * x + y * y + z * z;
    }
    __syncthreads();

    const int q = chunk * 256 + threadIdx.x;
    const float qx = sx[q], qy = sy[q], qz = sz[q], qxx = sxx[q];

    float bd[KNN];
    int   bi[KNN];
#pragma unroll
    for (int i = 0; i < KNN; ++i) { bd[i] = FMAX; bi[i] = 0; }

    for (int m = 0; m < NPTS; ++m) {
        float d = qxx + sxx[m] - 2.0f * (qx * sx[m] + qy * sy[m] + qz * sz[m]);
        if (d < bd[KNN - 1]) {
            // fully unrolled insertion network: constant indices, stays in VGPRs
#pragma unroll
            for (int i = KNN - 1; i >= 1; --i) {
                if (d < bd[i - 1])      { bd[i] = bd[i - 1]; bi[i] = bi[i - 1]; }
                else if (d < bd[i])     { bd[i] = d;         bi[i] = m;         }
            }
            if (d < bd[0])              { bd[0] = d;         bi[0] = m;         }
        }
    }
    int* op = idx + (size_t)(b * NPTS + q) * KNN;
#pragma unroll
    for (int i = 0; i < KNN; ++i) op[i] = bi[i];
}

// --------------------------------------------------------------------------
// Weight prep: Wcat (2O x Cpad) = rows [ Wa ; Wb - Wa ], zero-padded K cols.
// --------------------------------------------------------------------------
__global__ void pack_w_kernel(const float* __restrict__ W, float* __restrict__ Wc,
                              int O, int C, int Cpad)
{
    int t = blockIdx.x * blockDim.x + threadIdx.x;
    int total = 2 * O * Cpad;
    if (t >= total) return;
    int n = t / Cpad, k = t % Cpad;
    float v;
    if (k >= C)      v = 0.0f;
    else if (n < O)  v = W[n * (2 * C) + k];
    else             v = W[(n - O) * (2 * C) + C + k] - W[(n - O) * (2 * C) + k];
    Wc[t] = v;
}

// Layer-0 features are the true (B,N,3) rows of xyz; pad K-dim to 4.
__global__ void pack_xyz_kernel(const float* __restrict__ xyz, float* __restrict__ F0)
{
    int t = blockIdx.x * blockDim.x + threadIdx.x;
    if (t >= BATCH * NPTS) return;
    F0[t * 4 + 0] = xyz[t * 3 + 0];
    F0[t * 4 + 1] = xyz[t * 3 + 1];
    F0[t * 4 + 2] = xyz[t * 3 + 2];
    F0[t * 4 + 3] = 0.0f;
}

// --------------------------------------------------------------------------
// GEMM: G[m, 0:2O] = F[m, 0:CDIM] @ Wcat^T using V_WMMA_F32_16X16X4_F32.
// Each wave: 16(M) x 64(N) strip, 4 f32 accumulators share one A fragment.
// A frag (16x4): lanes 0-15 -> M=lane, K=0,1 ; lanes 16-31 -> M=lane-16, K=2,3.
// B frag (4x16): lanes 0-15 -> N=lane, K=0,1 ; lanes 16-31 -> N=lane-16, K=2,3.
// D (16x16 f32): VGPR v, lane<16 -> (M=v, N=lane); lane>=16 -> (M=v+8, N=lane-16).
// --------------------------------------------------------------------------
template <int CDIM>
__global__ __launch_bounds__(64) void gemm_wmma_kernel(const float* __restrict__ F,
                                                       const float* __restrict__ Wc,
                                                       float* __restrict__ G,
                                                       int twoO)
{
    const int lane = threadIdx.x & 31;
    const int wave = threadIdx.x >> 5;
    const int m0 = blockIdx.x * 16;
    const int n0 = (blockIdx.y * 2 + wave) * 64;
    const int nl = lane & 15;
    const int hi = lane >> 4;
    const int kb = hi * 2;

    const float* arow = F  + (size_t)(m0 + nl) * CDIM + kb;
    const float* b0p  = Wc + (size_t)(n0 +  0 + nl) * CDIM + kb;
    const float* b1p  = Wc + (size_t)(n0 + 16 + nl) * CDIM + kb;
    const float* b2p  = Wc + (size_t)(n0 + 32 + nl) * CDIM + kb;
    const float* b3p  = Wc + (size_t)(n0 + 48 + nl) * CDIM + kb;

    __builtin_prefetch(arow, 0, 0);   // global_prefetch_b8

    v8f acc0 = {}, acc1 = {}, acc2 = {}, acc3 = {};
#pragma unroll
    for (int k0 = 0; k0 < CDIM; k0 += 4) {
        v2f a  = *(const v2f*)(arow + k0);
        v2f b0 = *(const v2f*)(b0p  + k0);
        v2f b1 = *(const v2f*)(b1p  + k0);
        v2f b2 = *(const v2f*)(b2p  + k0);
        v2f b3 = *(const v2f*)(b3p  + k0);
        acc0 = __builtin_amdgcn_wmma_f32_16x16x4_f32(false, a, false, b0, (short)0, acc0, false, false);
        acc1 = __builtin_amdgcn_wmma_f32_16x16x4_f32(false, a, false, b1, (short)0, acc1, false, false);
        acc2 = __builtin_amdgcn_wmma_f32_16x16x4_f32(false, a, false, b2, (short)0, acc2, false, false);
        acc3 = __builtin_amdgcn_wmma_f32_16x16x4_f32(false, a, false, b3, (short)0, acc3, false, false);
    }

    float* gbase = G + (size_t)(m0 + hi * 8) * twoO + n0 + nl;
#pragma unroll
    for (int v = 0; v < 8; ++v) {
        float* gr = gbase + (size_t)v * twoO;
        gr[0]  = acc0[v];
        gr[16] = acc1[v];
        gr[32] = acc2[v];
        gr[48] = acc3[v];
    }
}

// --------------------------------------------------------------------------
// Edge aggregation: y_k = (G1[idx_k] + G2[n]) * s + b, leaky, max over k.
// One block per (b,n), one thread per output channel (coalesced G rows; G
// lives entirely in the 192 MB L2).
// --------------------------------------------------------------------------
__global__ void edge_max_kernel(const float* __restrict__ G,
                                const int* __restrict__ idx,
                                const float* __restrict__ scale,
                                const float* __restrict__ shift,
                                float* __restrict__ Fout,
                                int O)
{
    const int bn = blockIdx.x;
    const int b  = bn >> 11;
    const int o  = threadIdx.x;
    const int twoO = 2 * O;
    const int* ip = idx + (size_t)bn * KNN;
    const float g2 = G[(size_t)bn * twoO + O + o];
    const float s  = scale[o];
    const float sh = shift[o];
    float best = -FMAX;
#pragma unroll
    for (int k = 0; k < KNN; ++k) {
        int nk = ip[k];
        float g1 = G[(size_t)(b * NPTS + nk) * twoO + o];
        float y  = (g1 + g2) * s + sh;
        y = (y >= 0.0f) ? y : NEG_SLOPE * y;
        best = fmaxf(best, y);
    }
    Fout[(size_t)bn * O + o] = best;
}

// Final global max over N for layer 2 -> out[b*256 + o] (== (B,256,1) flat).
__global__ __launch_bounds__(256) void reduce_max_kernel(const float* __restrict__ Y,
                                                         float* __restrict__ out)
{
    const int b = blockIdx.x, o = threadIdx.x;
    float m = -FMAX;
    for (int n = 0; n < NPTS; ++n)
        m = fmaxf(m, Y[(size_t)(b * NPTS + n) * 256 + o]);
    out[b * 256 + o] = m;
}

// --------------------------------------------------------------------------
extern "C" void kernel_launch(void* const* d_in, const int* in_sizes, int n_in,
                              void* d_out, int out_size, void* d_ws, size_t ws_size,
                              hipStream_t stream)
{
    const float* xyz = (const float*)d_in[0];
    const float* W0  = (const float*)d_in[1];
    const float* s0  = (const float*)d_in[2];
    const float* b0  = (const float*)d_in[3];
    const float* W1  = (const float*)d_in[4];
    const float* s1  = (const float*)d_in[5];
    const float* b1  = (const float*)d_in[6];
    const float* W2  = (const float*)d_in[7];
    const float* s2  = (const float*)d_in[8];
    const float* b2  = (const float*)d_in[9];
    float* out = (float*)d_out;

    char* ws = (char*)d_ws;
    size_t off = 0;
    auto alloc = [&](size_t bytes) -> void* {
        void* p = ws + off;
        off = (off + bytes + 255) & ~(size_t)255;
        return p;
    };
    int*   idx = (int*)  alloc((size_t)BATCH * NPTS * KNN * 4);  //   2 MB
    float* F0  = (float*)alloc((size_t)BATCH * NPTS * 4   * 4);  //   1 MB
    float* Wc0 = (float*)alloc((size_t)128 * 4   * 4);
    float* Wc1 = (float*)alloc((size_t)256 * 64  * 4);
    float* Wc2 = (float*)alloc((size_t)512 * 128 * 4);
    float* G   = (float*)alloc((size_t)BATCH * NPTS * 512 * 4);  // 134 MB
    float* F1  = (float*)alloc((size_t)BATCH * NPTS * 64  * 4);  //  16 MB
    float* F2  = (float*)alloc((size_t)BATCH * NPTS * 128 * 4);  //  32 MB
    float* Y2  = (float*)alloc((size_t)BATCH * NPTS * 256 * 4);  //  64 MB

    const int BN = BATCH * NPTS;

    // prep
    pack_xyz_kernel<<<BN / 256, 256, 0, stream>>>(xyz, F0);
    pack_w_kernel<<<(2 * 64  * 4   + 255) / 256, 256, 0, stream>>>(W0, Wc0, 64, 3, 4);
    pack_w_kernel<<<(2 * 128 * 64  + 255) / 256, 256, 0, stream>>>(W1, Wc1, 128, 64, 64);
    pack_w_kernel<<<(2 * 256 * 128 + 255) / 256, 256, 0, stream>>>(W2, Wc2, 256, 128, 128);
    knn_kernel<<<BATCH * (NPTS / 256), 256, 0, stream>>>(xyz, idx);

    // layer 0: C=3(pad 4) -> O=64
    gemm_wmma_kernel<4><<<dim3(BN / 16, 128 / 128), 64, 0, stream>>>(F0, Wc0, G, 128);
    edge_max_kernel<<<BN, 64, 0, stream>>>(G, idx, s0, b0, F1, 64);

    // layer 1: C=64 -> O=128
    gemm_wmma_kernel<64><<<dim3(BN / 16, 256 / 128), 64, 0, stream>>>(F1, Wc1, G, 256);
    edge_max_kernel<<<BN, 128, 0, stream>>>(G, idx, s1, b1, F2, 128);

    // layer 2: C=128 -> O=256, then global max over N
    gemm_wmma_kernel<128><<<dim3(BN / 16, 512 / 128), 64, 0, stream>>>(F2, Wc2, G, 512);
    edge_max_kernel<<<BN, 256, 0, stream>>>(G, idx, s2, b2, Y2, 256);
    reduce_max_kernel<<<BATCH, 256, 0, stream>>>(Y2, out);
}